// JCIG_GCN_MLP_83004537962759
// MI455X (gfx1250) — compile-verified
//
#include <hip/hip_runtime.h>
#include <hip/hip_bf16.h>

typedef __attribute__((ext_vector_type(2))) float v2f;
typedef __attribute__((ext_vector_type(8))) float v8f;

#define N_NODES 50000
#define N_EDGES 800000
#define N_GRAPHS 64
#define IN_DIM 384
#define HID 128
#define MLP_HID 64

// ---------------------------------------------------------------------------
// Utility: zero a float region
// ---------------------------------------------------------------------------
__global__ void gcn_zero_kernel(float* __restrict__ p, int n) {
    int i = blockIdx.x * blockDim.x + threadIdx.x;
    if (i < n) p[i] = 0.0f;
}

// ---------------------------------------------------------------------------
// Degree accumulation over dst (A matrix), then dis = rsqrt(deg + 1)
// ---------------------------------------------------------------------------
__global__ void gcn_deg_kernel(const int* __restrict__ dst, float* __restrict__ deg, int E) {
    int e = blockIdx.x * blockDim.x + threadIdx.x;
    if (e < E) atomicAdd(&deg[dst[e]], 1.0f);
}

__global__ void gcn_rsqrt_kernel(float* __restrict__ dis, int n) {
    int i = blockIdx.x * blockDim.x + threadIdx.x;
    if (i < n) dis[i] = rsqrtf(dis[i] + 1.0f);
}

// ---------------------------------------------------------------------------
// WMMA f32 GEMM with fused GCN-normalization epilogue:
//   c  = A[M x K] @ W[K x 128]       (V_WMMA_F32_16X16X4_F32, exact f32 FMA)
//   hs  = c * dis[row]               (pre-scaled messages, gathered by edges)
//   agg = c * dis[row]^2             (self-loop contribution, atomics land here)
// One block = one 16-row M tile (A staged in LDS, padded stride K+1 to avoid
// the 16-way bank conflict of stride-K rows). 8 waves, wave w owns cols 16w..
// Saves a full 25.6 MB h-buffer round trip per layer vs. a separate scale pass.
// ---------------------------------------------------------------------------
__global__ __launch_bounds__(256) void gcn_gemm_wmma_f32(
    const float* __restrict__ A, const float* __restrict__ W,
    const float* __restrict__ dis,
    float* __restrict__ hs, float* __restrict__ agg, int M, int K)
{
    __shared__ float asmem[16 * (IN_DIM + 1)];   // max K = 384 -> 24640 B
    const int stride = K + 1;
    const int tid = threadIdx.x;
    const int m0  = blockIdx.x * 16;

    // Cooperative, coalesced load of the 16 x K A-tile into LDS.
    for (int idx = tid; idx < 16 * K; idx += 256) {
        int r = idx / K, c = idx - r * K;
        int gr = m0 + r;
        asmem[r * stride + c] = (gr < M) ? A[(size_t)gr * K + c] : 0.0f;
    }
    __syncthreads();

    const int wave = tid >> 5;       // 0..7 -> N tile index
    const int lane = tid & 31;
    const int half = lane >> 4;      // lane group: K offset 0 or 2
    const int r    = lane & 15;      // A: row M; B/D: column N
    const int col0 = wave * 16;

    v8f c = {};
    const float* wbase = W + col0 + r;        // B column N = col0 + r
    const float* arow  = asmem + r * stride;  // A row M = r

    for (int k = 0; k < K; k += 4) {
        const int ka = k + 2 * half;
        v2f a, b;
        a.x = arow[ka];                       // A VGPR0: K = 2*half
        a.y = arow[ka + 1];                   // A VGPR1: K = 2*half + 1
        b.x = wbase[(size_t)ka * HID];        // B VGPR0: K = 2*half
        b.y = wbase[(size_t)(ka + 1) * HID];  // B VGPR1: K = 2*half + 1
        // 8 args: (neg_a, A, neg_b, B, c_mod, C, reuse_a, reuse_b)
        c = __builtin_amdgcn_wmma_f32_16x16x4_f32(
                false, a, false, b, (short)0, c, false, false);
    }

    // D layout: VGPR v holds (M = v + 8*half, N = r).
    // dis[gm] is uniform across each 16-lane group -> broadcast load.
#pragma unroll
    for (int v = 0; v < 8; ++v) {
        int gm = m0 + v + 8 * half;
        if (gm < M) {
            float d  = dis[gm];
            float vh = c[v] * d;
            size_t o = (size_t)gm * HID + col0 + r;
            hs[o]  = vh;
            agg[o] = vh * d;
        }
    }
}

// ---------------------------------------------------------------------------
// Edge aggregation: agg[dst] += hs[src] * dis[dst]
// One wave per edge, each lane handles 4 channels (float4 gather, 4 f32 atomics).
// All traffic is L2-resident (hs/agg are 25.6 MB each << 192 MB L2).
// ---------------------------------------------------------------------------
__global__ void gcn_edge_scatter(const int* __restrict__ src, const int* __restrict__ dst,
                                 const float* __restrict__ dis, const float* __restrict__ hs,
                                 float* __restrict__ agg, int E) {
    int t = blockIdx.x * blockDim.x + threadIdx.x;
    int e = t >> 5;
    if (e >= E) return;
    int c = (t & 31) << 2;
    int s = src[e], d = dst[e];
    float dd = dis[d];
    const float4 v = *reinterpret_cast<const float4*>(hs + (size_t)s * HID + c);
    float* base = agg + (size_t)d * HID + c;
    atomicAdd(base + 0, v.x * dd);
    atomicAdd(base + 1, v.y * dd);
    atomicAdd(base + 2, v.z * dd);
    atomicAdd(base + 3, v.w * dd);
}

// ---------------------------------------------------------------------------
// act = relu(agg + b); optionally fuse the mean-pool scatter (last layer):
// pooled[batch[node]][c] += act   (saves a 25.6 MB re-read of act)
// ---------------------------------------------------------------------------
__global__ void gcn_finalize_kernel(const float* __restrict__ agg, const float* __restrict__ b,
                                    float* __restrict__ act, const int* __restrict__ batch,
                                    float* __restrict__ pooled, int do_pool, int total) {
    int i = blockIdx.x * blockDim.x + threadIdx.x;
    if (i >= total) return;
    int c = i & (HID - 1);
    float v = fmaxf(agg[i] + b[c], 0.0f);
    act[i] = v;
    if (do_pool) {
        int g = batch[i >> 7];
        atomicAdd(&pooled[(size_t)g * HID + c], v);
    }
}

// ---------------------------------------------------------------------------
// Global mean pool: node counts, then divide
// ---------------------------------------------------------------------------
__global__ void gcn_count_kernel(const int* __restrict__ batch, float* __restrict__ counts, int n) {
    int i = blockIdx.x * blockDim.x + threadIdx.x;
    if (i < n) atomicAdd(&counts[batch[i]], 1.0f);
}

__global__ void gcn_pool_div(float* __restrict__ pooled, const float* __restrict__ counts, int total) {
    int i = blockIdx.x * blockDim.x + threadIdx.x;
    if (i >= total) return;
    pooled[i] /= fmaxf(counts[i >> 7], 1.0f);
}

// ---------------------------------------------------------------------------
// MLP head: out[g] = relu(pooled[g] @ Wm1 + bm1) @ Wm2 + bm2
// One block per graph, one thread per hidden unit.
// ---------------------------------------------------------------------------
__global__ __launch_bounds__(MLP_HID) void gcn_mlp_head(
    const float* __restrict__ pooled, const float* __restrict__ Wm1,
    const float* __restrict__ bm1, const float* __restrict__ Wm2,
    const float* __restrict__ bm2, float* __restrict__ out)
{
    __shared__ float zsh[MLP_HID];
    int g = blockIdx.x, j = threadIdx.x;
    float acc = bm1[j];
    const float* p = pooled + (size_t)g * HID;
#pragma unroll 8
    for (int k = 0; k < HID; ++k) acc += p[k] * Wm1[k * MLP_HID + j];
    zsh[j] = fmaxf(acc, 0.0f) * Wm2[j];
    __syncthreads();
    for (int s = MLP_HID / 2; s > 0; s >>= 1) {
        if (j < s) zsh[j] += zsh[j + s];
        __syncthreads();
    }
    if (j == 0) out[g] = zsh[0] + bm2[0];
}

// ---------------------------------------------------------------------------
// Orchestration
// ---------------------------------------------------------------------------
extern "C" void kernel_launch(void* const* d_in, const int* in_sizes, int n_in,
                              void* d_out, int out_size, void* d_ws, size_t ws_size,
                              hipStream_t stream) {
    (void)in_sizes; (void)n_in; (void)out_size; (void)ws_size;

    const float* x     = (const float*)d_in[0];
    const int*   eidx  = (const int*)  d_in[1];   // [2, E]: src then dst
    const int*   batch = (const int*)  d_in[2];
    const float* W1 = (const float*)d_in[3];  const float* b1 = (const float*)d_in[4];
    const float* W2 = (const float*)d_in[5];  const float* b2 = (const float*)d_in[6];
    const float* W3 = (const float*)d_in[7];  const float* b3 = (const float*)d_in[8];
    const float* Wm1 = (const float*)d_in[9];  const float* bm1 = (const float*)d_in[10];
    const float* Wm2 = (const float*)d_in[11]; const float* bm2 = (const float*)d_in[12];

    const int* src = eidx;
    const int* dst = eidx + N_EDGES;

    // Workspace layout (floats): act | hs | agg | dis | pooled | counts
    float* ws = (float*)d_ws;
    const size_t NH = (size_t)N_NODES * HID;      // 6.4M floats
    float* act    = ws;                           // [N, 128] activations
    float* hs     = ws + NH;                      // [N, 128] scaled messages
    float* agg    = ws + 2 * NH;                  // [N, 128] aggregation target
    float* dis    = ws + 3 * NH;                  // [N] (deg first, then rsqrt)
    float* pooled = dis + N_NODES;                // [64, 128]
    float* counts = pooled + N_GRAPHS * HID;      // [64]

    const int B = 256;
    const int NH_i = (int)NH;

    // 1) zero deg + pooled + counts (contiguous region)
    {
        int nz = N_NODES + N_GRAPHS * HID + N_GRAPHS;
        gcn_zero_kernel<<<(nz + B - 1) / B, B, 0, stream>>>(dis, nz);
    }

    // 2) degree + dis = rsqrt(deg + 1)
    gcn_deg_kernel<<<(N_EDGES + B - 1) / B, B, 0, stream>>>(dst, dis, N_EDGES);
    gcn_rsqrt_kernel<<<(N_NODES + B - 1) / B, B, 0, stream>>>(dis, N_NODES);

    // 3) three GCN layers (GEMM epilogue emits hs + self-loop agg directly)
    const float* Win[3] = {W1, W2, W3};
    const float* bin[3] = {b1, b2, b3};
    const float* layer_in = x;
    int K = IN_DIM;
    for (int l = 0; l < 3; ++l) {
        gcn_gemm_wmma_f32<<<N_NODES / 16, 256, 0, stream>>>(
            layer_in, Win[l], dis, hs, agg, N_NODES, K);
        gcn_edge_scatter<<<(N_EDGES * 32 + B - 1) / B, B, 0, stream>>>(
            src, dst, dis, hs, agg, N_EDGES);
        gcn_finalize_kernel<<<(NH_i + B - 1) / B, B, 0, stream>>>(
            agg, bin[l], act, batch, pooled, (l == 2) ? 1 : 0, NH_i);
        layer_in = act;
        K = HID;
    }

    // 4) global mean pool (sums already fused into last finalize)
    gcn_count_kernel<<<(N_NODES + B - 1) / B, B, 0, stream>>>(batch, counts, N_NODES);
    gcn_pool_div<<<(N_GRAPHS * HID + B - 1) / B, B, 0, stream>>>(pooled, counts, N_GRAPHS * HID);

    // 5) MLP head
    gcn_mlp_head<<<N_GRAPHS, MLP_HID, 0, stream>>>(pooled, Wm1, bm1, Wm2, bm2, (float*)d_out);
}